// kernelLogLikelihood_17282948399820
// MI455X (gfx1250) — compile-verified
//
#include <hip/hip_runtime.h>
#include <hip/hip_bf16.h>
#include <math.h>

// Problem constants (from reference)
#define BN    4096
#define DIM   768
#define DC    100
#define NREF  50000
#define NCHUNK 4
#define CHUNKN (NREF / NCHUNK)   // 12500

typedef __attribute__((ext_vector_type(2))) float v2f;
typedef __attribute__((ext_vector_type(8))) float v8f;

// ---------------------------------------------------------------------------
// Kernel 1: z = x @ W^T + bias   (one block per row of x), plus zn = ||z||^2
// ---------------------------------------------------------------------------
__global__ __launch_bounds__(128) void proj_kernel(const float* __restrict__ x,
                                                   const float* __restrict__ W,
                                                   const float* __restrict__ bias,
                                                   float* __restrict__ z,
                                                   float* __restrict__ zn) {
  __shared__ float xs[DIM];
  __shared__ float red[128];
  const int row = blockIdx.x;
  const int tid = threadIdx.x;

  for (int k = tid; k < DIM; k += 128) xs[k] = x[row * DIM + k];
  __syncthreads();

  float zv = 0.0f;
  if (tid < DC) {
    const float4* wr = (const float4*)(W + tid * DIM);  // 3072B-aligned
    float acc = 0.0f;
#pragma unroll 4
    for (int t = 0; t < DIM / 4; ++t) {
      float4 w4 = wr[t];
      const float* xp = xs + 4 * t;
      acc = fmaf(w4.x, xp[0], acc);
      acc = fmaf(w4.y, xp[1], acc);
      acc = fmaf(w4.z, xp[2], acc);
      acc = fmaf(w4.w, xp[3], acc);
    }
    zv = acc + bias[tid];
    z[row * DC + tid] = zv;
  }
  red[tid] = zv * zv;
  __syncthreads();
#pragma unroll
  for (int off = 64; off > 0; off >>= 1) {
    if (tid < off) red[tid] += red[tid + off];
    __syncthreads();
  }
  if (tid == 0) zn[row] = red[0];
}

// ---------------------------------------------------------------------------
// Kernel 2: rn[j] = ||X_ref[j]||^2
// ---------------------------------------------------------------------------
__global__ __launch_bounds__(256) void rowsq_kernel(const float* __restrict__ Xref,
                                                    float* __restrict__ rn) {
  const int j = blockIdx.x * 256 + threadIdx.x;
  if (j >= NREF) return;
  const float4* p = (const float4*)(Xref + j * DC);  // 400B stride, 16B aligned
  float acc = 0.0f;
#pragma unroll
  for (int t = 0; t < DC / 4; ++t) {
    float4 v = p[t];
    acc = fmaf(v.x, v.x, acc);
    acc = fmaf(v.y, v.y, acc);
    acc = fmaf(v.z, v.z, acc);
    acc = fmaf(v.w, v.w, acc);
  }
  rn[j] = acc;
}

// ---------------------------------------------------------------------------
// Online logsumexp update (log2 domain) for one accumulator tile
// ---------------------------------------------------------------------------
__device__ __forceinline__ void lse_update(const v8f& acc, bool valid, float rc,
                                           const float* znc, float twoC2,
                                           float* m, float* s) {
  if (!valid) return;
#pragma unroll
  for (int r = 0; r < 8; ++r) {
    // log2-domain logit: (2*S - ||z||^2 - ||xr||^2) * c2, clamped (d2 >= 0)
    float t2 = fmaf(acc[r], twoC2, -znc[r]) - rc;
    t2 = fminf(t2, 0.0f);
    const float mOld = m[r];
    const float mNew = fmaxf(mOld, t2);
    s[r] = s[r] * __builtin_amdgcn_exp2f(mOld - mNew) +
           __builtin_amdgcn_exp2f(t2 - mNew);
    m[r] = mNew;
  }
}

// ---------------------------------------------------------------------------
// Kernel 3: fused  S = z @ X_ref^T  (f32 WMMA)  +  online logsumexp epilogue.
//   grid: (BN/16) x NCHUNK blocks, 256 threads (8 wave32).
//   Each wave owns TWO 16-column tiles per iteration (two independent WMMA
//   accumulator chains -> 2x ILP on the XDL pipe); each block sweeps a
//   12500-column chunk of X_ref and emits per-row (max,sum) partials.
// ---------------------------------------------------------------------------
__global__ __launch_bounds__(256) void kde_kernel(const float* __restrict__ z,
                                                  const float* __restrict__ zn,
                                                  const float* __restrict__ rn,
                                                  const float* __restrict__ Xref,
                                                  const float* __restrict__ hptr,
                                                  float* __restrict__ pm,
                                                  float* __restrict__ ps) {
  const int tid  = threadIdx.x;
  const int wave = tid >> 5;
  const int lane = tid & 31;
  const int h16  = lane >> 4;   // half-wave select (K phase / M phase)
  const int n16  = lane & 15;   // tile column (and A-row index)
  const int row0 = blockIdx.x * 16;
  const int chunk  = blockIdx.y;
  const int jstart = chunk * CHUNKN;
  const int jend   = jstart + CHUNKN;

  const float h     = hptr[0];
  const float LOG2E = 1.4426950408889634f;
  const float c2    = LOG2E / (2.0f * h * h);  // logits -> log2 domain scale
  const float twoC2 = 2.0f * c2;

  // --- preload A fragments: A[M=n16][K = 4t + 2*h16 + {0,1}] -----------------
  v2f afrag[25];
  const float* zrow = z + (row0 + n16) * DC + 2 * h16;
#pragma unroll
  for (int t = 0; t < 25; ++t) afrag[t] = *(const v2f*)(zrow + 4 * t);

  // per-lane ||z||^2 * c2 for the 8 rows this lane's accumulators cover
  float znc[8];
#pragma unroll
  for (int r = 0; r < 8; ++r) znc[r] = zn[row0 + r + 8 * h16] * c2;

  // per-lane online logsumexp state (log2 domain), one per accumulator row
  float m[8], s[8];
#pragma unroll
  for (int r = 0; r < 8; ++r) { m[r] = -INFINITY; s[r] = 0.0f; }

  const int niter = (CHUNKN + 255) / 256;  // 49
  for (int it = 0; it < niter; ++it) {
    const int base = jstart + it * 256 + wave * 32;
    const int j0 = base + n16;
    const int j1 = base + 16 + n16;
    const bool valid0 = (j0 < jend);
    const bool valid1 = (j1 < jend);
    const int jc0 = valid0 ? j0 : jstart;
    const int jc1 = valid1 ? j1 : jstart;
    const float* brow0 = Xref + jc0 * DC + 2 * h16;
    const float* brow1 = Xref + jc1 * DC + 2 * h16;

    // prefetch next iteration's rows (gfx1250 global_prefetch_b8)
    if (base + 256 + n16 < jend)
      __builtin_prefetch(Xref + (j0 + 256) * DC + 2 * h16, 0, 1);

    v8f acc0 = {};  // two independent accumulator chains
    v8f acc1 = {};
#pragma unroll
    for (int t = 0; t < 25; ++t) {
      v2f b0 = *(const v2f*)(brow0 + 4 * t);  // B[K=4t+2*h16+{0,1}][N=n16]
      v2f b1 = *(const v2f*)(brow1 + 4 * t);
      acc0 = __builtin_amdgcn_wmma_f32_16x16x4_f32(
          false, afrag[t], false, b0, (short)0, acc0, false, false);
      acc1 = __builtin_amdgcn_wmma_f32_16x16x4_f32(
          false, afrag[t], false, b1, (short)0, acc1, false, false);
    }

    lse_update(acc0, valid0, rn[jc0] * c2, znc, twoC2, m, s);
    lse_update(acc1, valid1, rn[jc1] * c2, znc, twoC2, m, s);
  }

  // --- merge the 16 column-slices within each half-wave ----------------------
#pragma unroll
  for (int r = 0; r < 8; ++r) {
#pragma unroll
    for (int msk = 1; msk < 16; msk <<= 1) {
      const float mo = __shfl_xor(m[r], msk, 32);
      const float so = __shfl_xor(s[r], msk, 32);
      const float mN = fmaxf(m[r], mo);
      s[r] = s[r] * __builtin_amdgcn_exp2f(m[r] - mN) +
             so   * __builtin_amdgcn_exp2f(mo   - mN);
      m[r] = mN;
    }
  }

  // --- merge the 8 waves through LDS ----------------------------------------
  __shared__ float lm[8][16];
  __shared__ float ls[8][16];
  if (n16 == 0) {
#pragma unroll
    for (int r = 0; r < 8; ++r) {
      const int row = r + 8 * h16;
      lm[wave][row] = m[r];
      ls[wave][row] = s[r];
    }
  }
  __syncthreads();

  if (tid < 16) {
    float M = lm[0][tid];
    float S = ls[0][tid];
#pragma unroll
    for (int w = 1; w < 8; ++w) {
      const float mo = lm[w][tid];
      const float so = ls[w][tid];
      const float mN = fmaxf(M, mo);
      S = S * __builtin_amdgcn_exp2f(M - mN) +
          so * __builtin_amdgcn_exp2f(mo - mN);
      M = mN;
    }
    pm[chunk * BN + row0 + tid] = M;
    ps[chunk * BN + row0 + tid] = S;
  }
}

// ---------------------------------------------------------------------------
// Kernel 4: merge the NCHUNK partials, convert back to natural log, add coeff
// ---------------------------------------------------------------------------
__global__ __launch_bounds__(256) void merge_kernel(const float* __restrict__ pm,
                                                    const float* __restrict__ ps,
                                                    const float* __restrict__ hptr,
                                                    float* __restrict__ out) {
  const int i = blockIdx.x * 256 + threadIdx.x;
  if (i >= BN) return;
  float M = pm[i];
  float S = ps[i];
#pragma unroll
  for (int c = 1; c < NCHUNK; ++c) {
    const float mo = pm[c * BN + i];
    const float so = ps[c * BN + i];
    const float mN = fmaxf(M, mo);
    S = S * __builtin_amdgcn_exp2f(M - mN) +
        so * __builtin_amdgcn_exp2f(mo - mN);
    M = mN;
  }
  const float h   = hptr[0];
  const float LN2 = 0.6931471805599453f;
  const float lse = (M + log2f(S)) * LN2;  // back to natural log
  const float coeff = -logf((float)NREF)
                      - logf(2.0f * 3.14159265358979323846f) * (0.5f * DC)
                      - logf(h) * (float)DC;
  out[i] = lse + coeff;
}

// ---------------------------------------------------------------------------
// Launch
// ---------------------------------------------------------------------------
extern "C" void kernel_launch(void* const* d_in, const int* in_sizes, int n_in,
                              void* d_out, int out_size, void* d_ws, size_t ws_size,
                              hipStream_t stream) {
  const float* x    = (const float*)d_in[0];  // [4096,768]
  const float* W    = (const float*)d_in[1];  // [100,768]
  const float* bias = (const float*)d_in[2];  // [100]
  const float* Xref = (const float*)d_in[3];  // [50000,100]
  const float* hbw  = (const float*)d_in[4];  // scalar
  float* out = (float*)d_out;                 // [4096]

  float* ws = (float*)d_ws;
  float* z  = ws;                  // 4096*100
  float* zn = z + BN * DC;         // 4096
  float* rn = zn + BN;             // 50000
  float* pm = rn + NREF;           // NCHUNK*4096 partial maxes
  float* ps = pm + NCHUNK * BN;    // NCHUNK*4096 partial sums   (total ~2.0 MB)

  proj_kernel<<<BN, 128, 0, stream>>>(x, W, bias, z, zn);
  rowsq_kernel<<<(NREF + 255) / 256, 256, 0, stream>>>(Xref, rn);
  dim3 grid(BN / 16, NCHUNK);
  kde_kernel<<<grid, 256, 0, stream>>>(z, zn, rn, Xref, hbw, pm, ps);
  merge_kernel<<<(BN + 255) / 256, 256, 0, stream>>>(pm, ps, hbw, out);
}